// candidate_creator_41764261986444
// MI455X (gfx1250) — compile-verified
//
#include <hip/hip_runtime.h>
#include <hip/hip_bf16.h>

// Problem constants (from reference)
#define BB 16
#define CC 4
#define FF 2048
#define TT 512
#define NUM_CANDS 4
#define MASKED_AREA 35

typedef __attribute__((ext_vector_type(2))) float v2f;
typedef __attribute__((ext_vector_type(8))) float v8f;

// ---------------------------------------------------------------------------
// Kernel 1: rowmax[b,c,f] = max over T of x[b,c,f,:]
// One wave32 per row of T=512 contiguous floats. float4 loads -> b128 vmem,
// fully coalesced (512B per wave per iteration). This is the 256MB streaming
// pass that dominates runtime (~11us at 23.3 TB/s).
// ---------------------------------------------------------------------------
__global__ void rowmax_kernel(const float* __restrict__ x,
                              float* __restrict__ rowmax) {
    const int waveId = (int)((blockIdx.x * blockDim.x + threadIdx.x) >> 5);
    const int lane   = (int)(threadIdx.x & 31);
    const float4* p = (const float4*)(x + (size_t)waveId * TT);
    float m = -3.402823466e+38f;
#pragma unroll
    for (int it = 0; it < 4; ++it) {
        float4 v = p[it * 32 + lane];
        m = fmaxf(m, fmaxf(fmaxf(v.x, v.y), fmaxf(v.z, v.w)));
    }
#pragma unroll
    for (int off = 16; off > 0; off >>= 1)
        m = fmaxf(m, __shfl_xor(m, off, 32));
    if (lane == 0) rowmax[waveId] = m;
}

// ---------------------------------------------------------------------------
// Kernel 2: per-(b,c) iterative peak picking with +/-35 frequency NMS.
// One 256-thread block per (b,c); each thread owns 8 of the F=2048 rowmax
// values. Masked frequencies contribute -100.0 (exactly like the reference's
// jnp.where). Tie-break: smaller f wins (matches flat argmax semantics).
// ---------------------------------------------------------------------------
__global__ void nms_kernel(const float* __restrict__ rowmax,
                           float* __restrict__ pooled,   // [NUM_CANDS*BB, CC]
                           int* __restrict__ freq0) {    // [NUM_CANDS*BB]
    const int bc = (int)blockIdx.x;       // b*CC + c
    const int b  = bc >> 2;
    const int c  = bc & 3;
    const float* row = rowmax + (size_t)bc * FF;

    const int base = (int)threadIdx.x * 8;
    float v[8];
#pragma unroll
    for (int j = 0; j < 8; ++j) v[j] = row[base + j];

    __shared__ float sval[256];
    __shared__ int   sidx[256];

    int centers[NUM_CANDS - 1];
    int ncent = 0;

    for (int cand = 0; cand < NUM_CANDS; ++cand) {
        float best  = -3.402823466e+38f;
        int   bestf = FF;
#pragma unroll
        for (int j = 0; j < 8; ++j) {
            const int f = base + j;
            bool masked = false;
            for (int k = 0; k < ncent; ++k) {
                int d = f - centers[k];
                if (d < 0) d = -d;
                masked |= (d < MASKED_AREA);
            }
            const float val = masked ? -100.0f : v[j];
            if (val > best || (val == best && f < bestf)) { best = val; bestf = f; }
        }
        sval[threadIdx.x] = best;
        sidx[threadIdx.x] = bestf;
        __syncthreads();
        for (int s = 128; s > 0; s >>= 1) {
            if ((int)threadIdx.x < s) {
                const float ov = sval[threadIdx.x + s];
                const int   oi = sidx[threadIdx.x + s];
                if (ov > sval[threadIdx.x] ||
                    (ov == sval[threadIdx.x] && oi < sidx[threadIdx.x])) {
                    sval[threadIdx.x] = ov;
                    sidx[threadIdx.x] = oi;
                }
            }
            __syncthreads();
        }
        const float pv = sval[0];
        const int   pf = sidx[0];
        if (threadIdx.x == 0) {
            pooled[(cand * BB + b) * CC + c] = pv;
            if (c == 0) freq0[cand * BB + b] = pf;
        }
        if (cand < NUM_CANDS - 1) centers[ncent++] = pf;
        __syncthreads();   // protect sval/sidx before next round
    }
}

// ---------------------------------------------------------------------------
// Kernel 3: logits = pooled[64,4] @ W[4,2] + b, plus periods gather.
// Uses V_WMMA_F32_16X16X4_F32: 4 waves, each computing one 16x16 tile
// (A = 16 pooled rows x K=4 channels; B = W padded to 4x16, cols>=2 are 0).
// Layouts per ISA 7.12.2:
//   A 16x4 f32: lanes 0-15 M=0..15 {K0,K1}; lanes 16-31 M=0..15 {K2,K3}
//   B 4x16 f32: lanes 0-15 N=0..15 {K0,K1}; lanes 16-31 N=0..15 {K2,K3}
//   D: VGPR r -> M=r (lanes 0-15) / M=8+r (lanes 16-31), N=lane%16
// EXEC is all-ones at the WMMA (divergence only afterwards).
// ---------------------------------------------------------------------------
__global__ void logits_kernel(const float* __restrict__ pooled,
                              const int* __restrict__ freq0,
                              const float* __restrict__ W,     // [4,2]
                              const float* __restrict__ bias,  // [2]
                              const float* __restrict__ periods, // [F]
                              float* __restrict__ out) {       // [64,3]
    const int wave = (int)(threadIdx.x >> 5);   // 0..3 -> rows 16w..16w+15
    const int lane = (int)(threadIdx.x & 31);
    const int half = lane >> 4;                 // 0: K=0,1   1: K=2,3
    const int l    = lane & 15;

    const int row = wave * 16 + l;              // A-matrix row (M=l)
    v2f a, bm;
    a.x = pooled[row * CC + (half * 2 + 0)];
    a.y = pooled[row * CC + (half * 2 + 1)];
    bm.x = (l < 2) ? W[(half * 2 + 0) * 2 + l] : 0.0f;   // B[k][n=l]
    bm.y = (l < 2) ? W[(half * 2 + 1) * 2 + l] : 0.0f;

    v8f c = {};
    c = __builtin_amdgcn_wmma_f32_16x16x4_f32(
        /*neg_a=*/false, a, /*neg_b=*/false, bm,
        /*c_mod=*/(short)0, c, /*reuse_a=*/false, /*reuse_b=*/false);

    if (l < 2) {
        const float bv = bias[l];
#pragma unroll
        for (int r = 0; r < 8; ++r) {
            const int m = half * 8 + r;                 // tile row
            out[(wave * 16 + m) * 3 + l] = c[r] + bv;   // logits column l
        }
    }
    if (half == 0) {                                    // periods column
        const int n = wave * 16 + l;
        out[n * 3 + 2] = periods[freq0[n]];
    }
}

// ---------------------------------------------------------------------------
extern "C" void kernel_launch(void* const* d_in, const int* in_sizes, int n_in,
                              void* d_out, int out_size, void* d_ws, size_t ws_size,
                              hipStream_t stream) {
    const float* x       = (const float*)d_in[0];  // [16,4,2048,512]
    const float* periods = (const float*)d_in[1];  // [2048]
    const float* W       = (const float*)d_in[2];  // [4,2]
    const float* bias    = (const float*)d_in[3];  // [2]
    float* out = (float*)d_out;                    // [64,3]

    // Workspace layout
    float* rowmax = (float*)d_ws;                              // 131072 f32 = 512 KB
    float* pooled = (float*)((char*)d_ws + (size_t)BB * CC * FF * sizeof(float)); // 256 f32
    int*   freq0  = (int*)(pooled + NUM_CANDS * BB * CC);      // 64 i32

    // 1) rowmax: one wave per row, 131072 rows, 8 waves/block
    {
        const int rows = BB * CC * FF;           // 131072
        const int threads = 256;                 // 8 waves
        const int blocks = rows / 8;             // 16384
        rowmax_kernel<<<blocks, threads, 0, stream>>>(x, rowmax);
    }
    // 2) NMS peak picking: one block per (b,c)
    nms_kernel<<<BB * CC, 256, 0, stream>>>(rowmax, pooled, freq0);
    // 3) WMMA epilogue: 4 waves in one block
    logits_kernel<<<1, 128, 0, stream>>>(pooled, freq0, W, bias, periods, out);
}